// ZonoAlphaReLU_42193758716132
// MI455X (gfx1250) — compile-verified
//
#include <hip/hip_runtime.h>
#include <cstddef>

// CDNA5 / gfx1250, wave32.
typedef __attribute__((ext_vector_type(2))) float v2f;
typedef __attribute__((ext_vector_type(8))) float v8f;

#define NROWS 256      // generator (eps) axis
#define NCOLS 8192     // 8 * 1024 flattened neuron axis

// ---------------------------------------------------------------------------
// Kernel 1: zero the new_eps_terms block (8192 x 8192 f32 = 256 MB).
// Pure bandwidth: float4 grid-stride stores (512 B per wave32 store).
// ---------------------------------------------------------------------------
__global__ void __launch_bounds__(256) zono_zero_eps(float4* __restrict__ p, long n4) {
  long i = (long)blockIdx.x * blockDim.x + threadIdx.x;
  long stride = (long)gridDim.x * blockDim.x;
  float4 z; z.x = 0.f; z.y = 0.f; z.z = 0.f; z.w = 0.f;
  for (; i < n4; i += stride) p[i] = z;
}

// ---------------------------------------------------------------------------
// Kernel 2: one wave32 per 16 columns.
//   Phase A: radius[j] = sum_k |G[k][j]| via 64x V_WMMA_F32_16X16X4_F32 with
//            an all-ones A matrix (column-sum; K-slot-permutation invariant).
//            C/D layout => every lane holds radius of its own column in acc[0].
//   Phase B: elementwise zonotope ReLU math, write new_center + diagonal.
//   Phase C: float4 stream-scale of the 256x16 generator tile; per-column
//            factors redistributed with wave32 shuffles.
// ---------------------------------------------------------------------------
__global__ void __launch_bounds__(256) zono_relu_kernel(
    const float* __restrict__ center,
    const float* __restrict__ gen,
    const float* __restrict__ slope_param,
    float* __restrict__ out) {
  const int lane = threadIdx.x & 31;
  const int wave = (int)((blockIdx.x * blockDim.x + threadIdx.x) >> 5);
  const int col  = lane & 15;   // N within the 16x16 tile
  const int half = lane >> 4;   // 0: lanes 0-15, 1: lanes 16-31
  const int j    = wave * 16 + col;

  // ---- Phase A: WMMA column-sum reduction over 256 rows (K=4 per op) ----
  v8f acc = {};
  v2f a; a.x = 1.0f; a.y = 1.0f;   // A = ones(16x4): D[m][n] = sum_k B[k][n]
  #pragma unroll 4
  for (int s = 0; s < 64; ++s) {
    const int r0 = 4 * s + half;                 // rows r0 and r0+2 this lane
    float g0 = __builtin_fabsf(gen[(size_t)r0 * NCOLS + j]);
    float g1 = __builtin_fabsf(gen[(size_t)(r0 + 2) * NCOLS + j]);
    v2f b; b.x = g0; b.y = g1;                   // 4 rows x 16 cols per wave
    acc = __builtin_amdgcn_wmma_f32_16x16x4_f32(
        /*neg_a=*/false, a, /*neg_b=*/false, b,
        /*c_mod=*/(short)0, acc, /*reuse_a=*/false, /*reuse_b=*/false);
  }
  // lanes 0-15: D[M=0][N=lane]; lanes 16-31: D[M=8][N=lane-16] -> same sum.
  const float radius = acc[0];

  // ---- Phase B: per-neuron zonotope ReLU transform ----
  const float c  = center[j];
  const float sp = slope_param[j];
  const float slope = fminf(fmaxf(sp, 0.0f), 1.0f);
  const float u = c + radius;
  const float l = c - radius;
  const bool neg   = (u < 0.0f);
  const bool cross = (l < 0.0f) && (u > 0.0f);
  const float cf = cross ? 1.0f : 0.0f;
  const float thr = u / (u - l);                 // NaN/inf safe: cf kills it
  const float ng0 = (1.0f - slope) * u * 0.5f * cf;
  const float ng1 = -slope * l * 0.5f * cf;
  const float ngen = (slope <= thr) ? ng0 : ng1; // NaN compare -> ng1 (== 0)
  float ncen = cross ? (c * slope + ngen) : c;
  if (neg) ncen = 0.0f;
  const float factor = neg ? 0.0f : (cross ? slope : 1.0f);

  float* out_center = out;                                    // 8192
  float* out_gen    = out + NCOLS;                            // 256*8192
  float* out_eps    = out + NCOLS + (size_t)NROWS * NCOLS;    // 8192*8192

  if (half == 0) {
    out_center[j] = ncen;
    out_eps[(size_t)j * (NCOLS + 1)] = ngen;    // diagonal scatter
  }

  // ---- Phase C: float4 stream-scale (8 rows x 16 cols per wave-iter) ----
  // Lane l handles float4 column-group (lane&3) and rows k0 + (lane>>2).
  // Per-column factors live in lanes 0..15 -> broadcast via wave32 shuffles.
  const int g4 = lane & 3;
  const float f0 = __shfl(factor, g4 * 4 + 0, 32);
  const float f1 = __shfl(factor, g4 * 4 + 1, 32);
  const float f2 = __shfl(factor, g4 * 4 + 2, 32);
  const float f3 = __shfl(factor, g4 * 4 + 3, 32);

  const float4* __restrict__ gen4  = (const float4*)gen;
  float4* __restrict__       outg4 = (float4*)out_gen;
  const int c4 = wave * 4 + g4;                  // float4 column index
  for (int k = (lane >> 2); k < NROWS; k += 8) {
    const size_t off = (size_t)k * (NCOLS / 4) + c4;
    if (k + 32 < NROWS)
      __builtin_prefetch(&gen4[off + (size_t)32 * (NCOLS / 4)], 0, 1);
    float4 g = gen4[off];
    float4 r;
    r.x = g.x * f0; r.y = g.y * f1; r.z = g.z * f2; r.w = g.w * f3;
    outg4[off] = r;
  }
}

// ---------------------------------------------------------------------------
extern "C" void kernel_launch(void* const* d_in, const int* in_sizes, int n_in,
                              void* d_out, int out_size, void* d_ws, size_t ws_size,
                              hipStream_t stream) {
  const float* center = (const float*)d_in[0];   // (8,1024)
  const float* gen    = (const float*)d_in[1];   // (256,8,1024)
  const float* slope  = (const float*)d_in[2];   // (8,1024)
  float* out = (float*)d_out;

  // 1) Zero the 8192x8192 eps block. Offset (8192 + 256*8192)*4 B is 16B-aligned.
  float4* eps4 = (float4*)(out + NCOLS + (size_t)NROWS * NCOLS);
  const long n4 = ((long)NCOLS * NCOLS) / 4;     // 16,777,216 float4 stores
  zono_zero_eps<<<4096, 256, 0, stream>>>(eps4, n4);

  // 2) Main kernel: 512 waves (one per 16 columns), 8 waves per 256-thread block.
  const int waves = NCOLS / 16;                  // 512
  const int threads = 256;
  const int blocks = (waves * 32) / threads;     // 64
  zono_relu_kernel<<<blocks, threads, 0, stream>>>(center, gen, slope, out);
}